// AlleleEmbedding_16363825398339
// MI455X (gfx1250) — compile-verified
//
#include <hip/hip_runtime.h>

// ---------------------------------------------------------------------------
// AlleleEmbedding on MI455X (gfx1250, wave32).
// out[b,p,:] = (allele_table[a0] + allele_table[a1]) @ kernel_table[pos] + bias_table[pos]
// Memory-bound (~0.5 FLOP/byte, ~680 MB of gathers -> ~29 us floor at 23.3 TB/s).
// One wave per (b,p) site; matvec on V_WMMA_F32_16X16X4_F32.
// K matrix streamed in 1KB chunks via GLOBAL_LOAD_ASYNC_TO_LDS_B128, quad-buffered
// with prefetch depth 2; explicit s_wait_dscnt guard closes the async-write vs
// ds_read WAR race (async LDS writes are unordered w.r.t. DS ops per the ISA).
// ---------------------------------------------------------------------------

typedef float v2f __attribute__((ext_vector_type(2)));
typedef float v8f __attribute__((ext_vector_type(8)));

#define DDIM 64
#define WPB  8  // waves per block

template <int N>
__device__ __forceinline__ void wait_asynccnt() {
    asm volatile("s_wait_asynccnt %0" ::"n"(N) : "memory");
}
template <int N>
__device__ __forceinline__ void wait_dscnt() {
    asm volatile("s_wait_dscnt %0" ::"n"(N) : "memory");
}

// Async copy 16B/lane global -> LDS (ASYNCcnt-tracked, no VGPR round-trip).
// NOTE (CDNA5 ISA): INST_OFFSET is added to BOTH the LDS and the
// global address, so the same operands + offset:512 move the second half-chunk.
__device__ __forceinline__ void async_g2l_b128(unsigned lds_byte, const float* g) {
    asm volatile("global_load_async_to_lds_b128 %0, %1, off" ::"v"(lds_byte), "v"(g) : "memory");
}
__device__ __forceinline__ void async_g2l_b128_o512(unsigned lds_byte, const float* g) {
    asm volatile("global_load_async_to_lds_b128 %0, %1, off offset:512" ::"v"(lds_byte), "v"(g) : "memory");
}

__global__ __launch_bounds__(256) void allele_embed_wmma_kernel(
    const int*   __restrict__ alleles,       // [N,2]
    const int*   __restrict__ positions,     // [N]
    const float* __restrict__ allele_table,  // [16,64]
    const float* __restrict__ kernel_table,  // [NPOS,64*64] row-major [d][e]
    const float* __restrict__ bias_table,    // [NPOS,64]
    float*       __restrict__ out,           // [N,64]
    int N)
{
    __shared__ float lds_a[WPB][DDIM];
    __shared__ __align__(16) float kbuf[WPB][4][4 * DDIM];  // quad-buffered 1KB K-chunks

    const int lane = threadIdx.x & 31;
    const int wave = threadIdx.x >> 5;
    const int site = blockIdx.x * WPB + wave;  // one wave per (b,p); wave-uniform guard
    if (site >= N) return;

    const int i0  = alleles[site * 2 + 0];
    const int i1  = alleles[site * 2 + 1];
    const int pos = positions[site];

    // --- a = allele_table[i0] + allele_table[i1]  (64 floats, 2 per lane) ---
    const float a_lo = allele_table[i0 * DDIM + lane]      + allele_table[i1 * DDIM + lane];
    const float a_hi = allele_table[i0 * DDIM + lane + 32] + allele_table[i1 * DDIM + lane + 32];
    lds_a[wave][lane]      = a_lo;
    lds_a[wave][lane + 32] = a_hi;

    // --- 16 A fragments (16x4 fp32), chunk c = K rows 4c..4c+3, replicated over M.
    // A layout (ISA 7.12.2): lanes 0-15 hold K=0,1; lanes 16-31 hold K=2,3.
    const int j0 = (lane < 16) ? 0 : 2;
    v2f afrag[16];
#pragma unroll
    for (int c = 0; c < 16; ++c) {
        v2f f;
        f.x = lds_a[wave][4 * c + j0 + 0];
        f.y = lds_a[wave][4 * c + j0 + 1];
        afrag[c] = f;
    }

    const float* __restrict__ Kbase = kernel_table + (size_t)pos * (DDIM * DDIM);
    const int n = lane & 15;  // output column within a 16-wide tile

    // Issue async copy of chunk c (rows 4c..4c+3 = contiguous 1KB) into buffer c&3.
    auto issue_chunk = [&](int c) {
        const float* g = Kbase + (c << 8) + (lane << 2);                       // 16B per lane
        unsigned l = (unsigned)(uintptr_t)&kbuf[wave][c & 3][0] + (lane << 4); // LDS byte addr
        async_g2l_b128(l, g);
        async_g2l_b128_o512(l, g);
    };

    v8f acc[4];
#pragma unroll
    for (int t = 0; t < 4; ++t) acc[t] = (v8f){0.f, 0.f, 0.f, 0.f, 0.f, 0.f, 0.f, 0.f};

    // Prefetch depth 2: chunks 0 and 1 in flight before the MAC loop.
    issue_chunk(0);
    issue_chunk(1);

#pragma unroll
    for (int c = 0; c < 16; ++c) {
        if (c <= 13) {
            // WAR guard: buffer (c+2)&3 was last read by chunk c-2. DS completes
            // in order, so allowing only the newest 4 DS ops (chunk c-1's reads)
            // outstanding proves chunk c-2's reads have retired.
            wait_dscnt<4>();
            issue_chunk(c + 2);
            wait_asynccnt<4>();     // chunks c+1, c+2 may remain outstanding
        } else if (c == 14) {
            wait_asynccnt<2>();     // only chunk 15 outstanding
        } else {
            wait_asynccnt<0>();
        }
        const float* kb = &kbuf[wave][c & 3][0];
#pragma unroll
        for (int t = 0; t < 4; ++t) {
            // B (4x16) frag: lanes 0-15 rows {0,1}, lanes 16-31 rows {2,3} of the chunk.
            v2f b;
            b.x = kb[(j0 + 0) * DDIM + t * 16 + n];
            b.y = kb[(j0 + 1) * DDIM + t * 16 + n];
            acc[t] = __builtin_amdgcn_wmma_f32_16x16x4_f32(
                /*neg_a=*/false, afrag[c],
                /*neg_b=*/false, b,
                /*c_mod=*/(short)0, acc[t],
                /*reuse_a=*/false, /*reuse_b=*/false);
        }
    }

    // Every M-row of acc[t] equals out[16t+n]; lanes 0-15 carry row 0.
#pragma unroll
    for (int t = 0; t < 4; ++t) {
        if (lane < 16) {
            const int col = t * 16 + lane;
            out[(size_t)site * DDIM + col] = acc[t][0] + bias_table[(size_t)pos * DDIM + col];
        }
    }
}

extern "C" void kernel_launch(void* const* d_in, const int* in_sizes, int n_in,
                              void* d_out, int out_size, void* d_ws, size_t ws_size,
                              hipStream_t stream) {
    (void)n_in; (void)out_size; (void)d_ws; (void)ws_size;
    const int*   alleles      = (const int*)d_in[0];    // [B,P,2] int32
    const int*   positions    = (const int*)d_in[1];    // [B,P]   int32
    const float* allele_table = (const float*)d_in[2];  // [16,64]
    const float* kernel_table = (const float*)d_in[3];  // [NPOS,4096]
    const float* bias_table   = (const float*)d_in[4];  // [NPOS,64]
    float*       out          = (float*)d_out;          // [B,P,64]

    const int N = in_sizes[1];  // B*P sites
    const int blocks = (N + WPB - 1) / WPB;

    allele_embed_wmma_kernel<<<blocks, WPB * 32, 0, stream>>>(
        alleles, positions, allele_table, kernel_table, bias_table, out, N);
}